// CustomLoss_40097814676083
// MI455X (gfx1250) — compile-verified
//
#include <hip/hip_runtime.h>

// ---------------------------------------------------------------------------
// Soft-label cross-entropy loss, B=2097152 rows x C=18 classes, fp32.
// Memory-bound: ~168 MB / 23.3 TB/s ~= 7.2 us floor. Design:
//   Kernel 1: grid-stride streaming; 9x b64 loads per row (72B rows are
//             8B-aligned), branchless x[t]/x[n] selection, v_exp/v_log,
//             wave32 shfl-xor + LDS block reduction -> 2048 partials in d_ws.
//   Kernel 2: single wave32 folds 2048 partials with V_WMMA_F32_16X16X4_F32
//             (A = partials, B = ones => D rows = chunk sums), deterministic.
// ---------------------------------------------------------------------------

#define NBLOCKS  2048
#define NTHREADS 256
#define NCLS     18

typedef float v2f __attribute__((ext_vector_type(2)));
typedef float v8f __attribute__((ext_vector_type(8)));

__global__ __launch_bounds__(NTHREADS) void loss_partial_kernel(
    const float* __restrict__ outputs,
    const int*   __restrict__ targets,
    const int*   __restrict__ ages,
    float*       __restrict__ partial,
    int B)
{
    const int tid    = blockIdx.x * blockDim.x + threadIdx.x;
    const int stride = gridDim.x * blockDim.x;

    float acc = 0.0f;

    for (int r = tid; r < B; r += stride) {
        // Prefetch next grid-stride row (speculative; OOB silently dropped).
        __builtin_prefetch(outputs + (size_t)(r + stride) * NCLS, 0, 3);

        // --- age bands -> (delta, shift); integer ages => disjoint bands ---
        const int t = targets[r];
        const int a = ages[r];
        float delta; int shift;
        if      (a > 50 && a < 60) { delta = (float)(a - 50) * 0.05f; shift =  1; }
        else if (a == 60)          { delta = 0.2f;                    shift = -1; }
        else if (a > 24 && a < 30) { delta = (float)(a - 20) * 0.05f; shift =  1; }
        else if (a > 29 && a < 35) { delta = (float)(39 - a) * 0.05f; shift = -1; }
        else                       { delta = 0.0f;                    shift =  0; }
        int nb = t + shift;                       // wrap -1 -> 17, 18 -> 0
        nb = (nb < 0) ? (NCLS - 1) : ((nb >= NCLS) ? 0 : nb);

        // --- load one 72-byte row as 9 x b64 (always 8B aligned) ---
        const float2* rp = reinterpret_cast<const float2*>(outputs + (size_t)r * NCLS);
        float x[NCLS];
        #pragma unroll
        for (int j = 0; j < NCLS / 2; ++j) {
            float2 v = rp[j];
            x[2 * j]     = v.x;
            x[2 * j + 1] = v.y;
        }

        // --- logsumexp + branchless gather of x[t], x[nb] (no scratch) ---
        float m = x[0];
        #pragma unroll
        for (int j = 1; j < NCLS; ++j) m = fmaxf(m, x[j]);

        float s = 0.0f, xt = 0.0f, xn = 0.0f;
        #pragma unroll
        for (int j = 0; j < NCLS; ++j) {
            s += __expf(x[j] - m);                // v_exp_f32 (trans pipe)
            xt = (j == t)  ? x[j] : xt;           // v_cndmask chains
            xn = (j == nb) ? x[j] : xn;
        }
        const float lse = m + __logf(s);

        acc += lse - xt + delta * (xt - xn);
    }

    // --- wave32 reduction, then cross-wave via LDS ---
    #pragma unroll
    for (int off = 16; off > 0; off >>= 1)
        acc += __shfl_xor(acc, off, 32);

    __shared__ float wsum[NTHREADS / 32];
    const int lane = threadIdx.x & 31;
    const int wave = threadIdx.x >> 5;
    if (lane == 0) wsum[wave] = acc;
    __syncthreads();

    if (threadIdx.x == 0) {
        float b = 0.0f;
        #pragma unroll
        for (int i = 0; i < NTHREADS / 32; ++i) b += wsum[i];
        partial[blockIdx.x] = b;   // every block writes -> d_ws fully defined
    }
}

// Final reduction: one wave32. Fold 64 partials per V_WMMA_F32_16X16X4_F32:
// D(16x16) = A(16x4) x ones(4x16) + C  =>  D[m][n] = sum_k A[m][k].
// Column n of the accumulator lives in lane n (M=0..7, 8 VGPRs) and lane
// n+16 (M=8..15), so per-lane VGPR sum + shfl_xor(16) yields the total.
__global__ __launch_bounds__(32) void reduce_wmma_kernel(
    const float* __restrict__ partial,
    float*       __restrict__ out,
    float invB)
{
    const int lane = threadIdx.x;   // 0..31, full EXEC (WMMA requirement)

    v8f c = {};                     // f32 accumulator (8 VGPRs)
    v2f bones; bones[0] = 1.0f; bones[1] = 1.0f;   // B = all-ones 4x16

    for (int base = 0; base < NBLOCKS; base += 64) {
        v2f a;                                      // A = 64 partials (16x4)
        a[0] = partial[base + lane];
        a[1] = partial[base + 32 + lane];
        c = __builtin_amdgcn_wmma_f32_16x16x4_f32(
                /*neg_a=*/false, a, /*neg_b=*/false, bones,
                /*c_mod=*/(short)0, c, /*reuse_a=*/false, /*reuse_b=*/false);
    }

    float col = 0.0f;
    #pragma unroll
    for (int i = 0; i < 8; ++i) col += c[i];        // M=0..7 (or 8..15)
    col += __shfl_xor(col, 16, 32);                 // add other half of column

    if (lane == 0) out[0] = col * invB;             // mean over B
}

extern "C" void kernel_launch(void* const* d_in, const int* in_sizes, int n_in,
                              void* d_out, int out_size, void* d_ws, size_t ws_size,
                              hipStream_t stream)
{
    const float* outputs = (const float*)d_in[0];
    const int*   targets = (const int*)d_in[1];
    const int*   ages    = (const int*)d_in[2];
    const int    B       = in_sizes[1];

    float* partial = (float*)d_ws;   // NBLOCKS floats of scratch

    loss_partial_kernel<<<NBLOCKS, NTHREADS, 0, stream>>>(
        outputs, targets, ages, partial, B);

    reduce_wmma_kernel<<<1, 32, 0, stream>>>(
        partial, (float*)d_out, 1.0f / (float)B);
}